// GRU4Rec_67894843015522
// MI455X (gfx1250) — compile-verified
//
#include <hip/hip_runtime.h>
#include <hip/hip_bf16.h>

typedef __attribute__((ext_vector_type(2))) float v2f;
typedef __attribute__((ext_vector_type(8))) float v8f;

#define B_  4096
#define T_  200
#define H_  64
#define G_  192           // 3*H
#define XS  68            // LDS row stride for X and H tiles (bank-conflict-free A reads)
#define GS  196           // LDS row stride for GI tile (disjoint banks for M and M+8 halves)

__global__ __launch_bounds__(384) void gru_fused_wmma_f32(
    const int*  __restrict__ seq_token,  // [B, T]
    const int*  __restrict__ seq_pos,    // [B]
    const float* __restrict__ emb,       // [V, H]
    const float* __restrict__ W_ih,      // [3H, H]
    const float* __restrict__ W_hh,      // [3H, H]
    float* __restrict__ out)             // [B, H]
{
    __shared__ float Xs[2][16 * XS];     // double-buffered x_t tile, 16 x 64
    __shared__ float Hs[16 * XS];        // hidden state tile, 16 x 64
    __shared__ float GIs[16 * GS];       // r,z: gi+gh summed; n: gi only (cols 128..191)
    __shared__ float GHn[16 * XS];       // gh for the n-gate tile, 16 x 64
    __shared__ int   toks[16 * T_];      // all tokens for this batch tile (one burst)
    __shared__ int   lens[16];

    const int tid   = threadIdx.x;
    const int lane  = tid & 31;
    const int w     = tid >> 5;          // wave id 0..11 -> gate columns 16w..16w+15
    const int rb    = blockIdx.x * 16;   // first batch row of this tile

    // ---- Resident B-fragments: 16 rows of each weight matrix per wave ----
    // B (4x16) layout mirrors ISA A 16x4: VGPR0 = K k0+0 / k0+2, VGPR1 = k0+1 / k0+3.
    const int gcol  = 16 * w + (lane & 15);
    const int khalf = (lane >> 4) << 1;  // 0 or 2
    v2f bih[16], bhh[16];
#pragma unroll
    for (int j = 0; j < 16; ++j) {
        const int k = 4 * j + khalf;
        bih[j] = *(const v2f*)(W_ih + gcol * H_ + k);
        bhh[j] = *(const v2f*)(W_hh + gcol * H_ + k);
    }

    // ---- One-time staging: tokens burst, h=0, lengths ----
    for (int e = tid; e < 16 * T_; e += 384) toks[e] = seq_token[rb * T_ + e];
    for (int e = tid; e < 16 * XS; e += 384) Hs[e] = 0.0f;
    if (tid < 16) {
        int sp = seq_pos[rb + tid];
        lens[tid] = sp < 1 ? 1 : (sp > T_ ? T_ : sp);
    }
    __syncthreads();

    const int gm = tid >> 4;             // gather row (tid < 256)
    const int gq = tid & 15;             // gather quarter-row
    // ---- Prologue: gather x_0 ----
    if (tid < 256) {
        const int tok = toks[gm * T_];
        const float4 vv = ((const float4*)emb)[tok * (H_ / 4) + gq];
        *(float4*)(&Xs[0][gm * XS + gq * 4]) = vv;
    }
    __syncthreads();

    const int am   = lane & 15;          // A-matrix row
    const int row0 = (lane >> 4) << 3;   // C/D rows v / v+8
    const int ccol = 16 * w + (lane & 15);

    for (int t = 0; t < T_; ++t) {
        const int cur = t & 1;
        const int nxt = cur ^ 1;

        // ---- Pipelined gather: issue x_{t+1} load now, land it after the GEMM ----
        const bool pf = (tid < 256) && (t + 1 < T_);
        float4 pv;
        if (pf) {
            const int tok = toks[gm * T_ + t + 1];
            pv = ((const float4*)emb)[tok * (H_ / 4) + gq];
        }

        // ---- Preload all A-fragments, then two interleaved fp32 WMMA chains ----
        v2f ax[16], ah[16];
#pragma unroll
        for (int j = 0; j < 16; ++j) {
            const int k = 4 * j + khalf;
            ax[j] = *(const v2f*)(&Xs[cur][am * XS + k]);
            ah[j] = *(const v2f*)(&Hs[am * XS + k]);
        }
        v8f cgi = {};
        v8f cgh = {};
#pragma unroll
        for (int j = 0; j < 16; ++j) {
            cgi = __builtin_amdgcn_wmma_f32_16x16x4_f32(
                      false, ax[j], false, bih[j], (short)0, cgi, false, false);
            cgh = __builtin_amdgcn_wmma_f32_16x16x4_f32(
                      false, ah[j], false, bhh[j], (short)0, cgh, false, false);
        }

        // ---- Spill C/D tiles: r,z waves store gi+gh; n waves store both parts ----
        if (w < 8) {
#pragma unroll
            for (int v = 0; v < 8; ++v)
                GIs[(row0 + v) * GS + ccol] = cgi[v] + cgh[v];
        } else {
#pragma unroll
            for (int v = 0; v < 8; ++v) {
                GIs[(row0 + v) * GS + ccol] = cgi[v];
                GHn[(row0 + v) * XS + (ccol - 128)] = cgh[v];
            }
        }
        if (pf) *(float4*)(&Xs[nxt][gm * XS + gq * 4]) = pv;
        __syncthreads();

        // ---- Gate math + masked state update (1024 elems / 384 threads) ----
        for (int e = tid; e < 16 * H_; e += 384) {
            const int m = e >> 6;
            const int j = e & 63;
            const float sr = GIs[m * GS + j];
            const float sz = GIs[m * GS + 64 + j];
            const float r  = 1.0f / (1.0f + __expf(-sr));
            const float z  = 1.0f / (1.0f + __expf(-sz));
            const float a  = GIs[m * GS + 128 + j] + r * GHn[m * XS + j];
            const float n  = 1.0f - 2.0f / (__expf(2.0f * a) + 1.0f);   // tanh(a)
            const float h0 = Hs[m * XS + j];
            const float h1 = (1.0f - z) * n + z * h0;
            Hs[m * XS + j] = (t < lens[m]) ? h1 : h0;   // pack_padded: freeze past length
        }
        __syncthreads();
    }

    // ---- Epilogue: write final hidden state ----
    if (tid < 256) {
        const float4 vv = *(const float4*)(&Hs[gm * XS + gq * 4]);
        ((float4*)out)[(rb + gm) * (H_ / 4) + gq] = vv;
    }
}

extern "C" void kernel_launch(void* const* d_in, const int* in_sizes, int n_in,
                              void* d_out, int out_size, void* d_ws, size_t ws_size,
                              hipStream_t stream) {
    const int*   seq_token = (const int*)d_in[0];
    const int*   seq_pos   = (const int*)d_in[1];
    const float* emb       = (const float*)d_in[2];
    const float* W_ih      = (const float*)d_in[3];
    const float* W_hh      = (const float*)d_in[4];
    float*       out       = (float*)d_out;
    (void)in_sizes; (void)n_in; (void)out_size; (void)d_ws; (void)ws_size;

    dim3 grid(B_ / 16);
    dim3 block(384);
    gru_fused_wmma_f32<<<grid, block, 0, stream>>>(seq_token, seq_pos, emb, W_ih, W_hh, out);
}